// Head_78262894068063
// MI455X (gfx1250) — compile-verified
//
#include <hip/hip_runtime.h>
#include <hip/hip_bf16.h>

// ---------------------------------------------------------------------------
// Single-head causal attention, fp32, via V_WMMA_F32_16X16X4_F32 (CDNA5).
// B=8, T=2048, C=1024, H=64.
//   Kernel 1: Q,K,V = x @ Wq/Wk/Wv              (GEMM M=16384, N=64, K=1024)
//   Kernel 2: flash attention with online softmax, WMMA for QK^T and P@V.
// ---------------------------------------------------------------------------

typedef __attribute__((ext_vector_type(2))) float v2f;
typedef __attribute__((ext_vector_type(8))) float v8f;

static constexpr int BB = 8;      // batch
static constexpr int TT = 2048;   // sequence
static constexpr int CD = 1024;   // channels
static constexpr int HD = 64;     // head dim
static constexpr int BT = BB * TT;

static constexpr int MTILE = 128; // rows per block (proj) / queries per block (attn)
static constexpr int KC    = 32;  // K-chunk (proj)
static constexpr int XPAD  = 34;  // padded LDS stride for 32-wide chunks
static constexpr int SK    = 64;  // keys per chunk (attn)
static constexpr int KPAD  = 66;  // padded LDS stride for 64-wide chunks

// D(16x16) += A(16x4) * B(4x16), all f32. 8-arg form like the f16/bf16 WMMA.
__device__ __forceinline__ v8f wmma_f32(v2f a, v2f b, v8f c) {
    return __builtin_amdgcn_wmma_f32_16x16x4_f32(
        /*neg_a=*/false, a, /*neg_b=*/false, b,
        /*c_mod=*/(short)0, c, /*reuse_a=*/false, /*reuse_b=*/false);
}

// ---------------------------------------------------------------------------
// Kernel 1: fused Q/K/V projection.  grid = BT/128 blocks x 256 threads.
// Wave w computes rows [blk*128 + 16w, +16) of Q, K and V (16x64 each).
// ---------------------------------------------------------------------------
__global__ __launch_bounds__(256) void proj_qkv_kernel(
    const float* __restrict__ x,
    const float* __restrict__ Wq, const float* __restrict__ Wk,
    const float* __restrict__ Wv,
    float* __restrict__ Q, float* __restrict__ K, float* __restrict__ V)
{
    __shared__ float xs[MTILE * XPAD];      // x chunk, row-major [128][32] (+pad)
    __shared__ float ws[3][HD * XPAD];      // W chunks TRANSPOSED: [n][k] (+pad)

    const int tid  = threadIdx.x;
    const int wave = tid >> 5;
    const int lane = tid & 31;
    const int grp  = lane >> 4;   // 0/1 : which 16-lane half
    const int idx  = lane & 15;
    const long rowBlock = (long)blockIdx.x * MTILE;

    const float* Wm[3] = {Wq, Wk, Wv};

    v8f acc[3][4] = {};           // [qkv][n-tile] 16x16 f32 accumulators

    for (int kc = 0; kc < CD; kc += KC) {
        __syncthreads();
        // --- stage x chunk: 128 rows x 32 cols (float4 loads) ---
        for (int i = tid; i < MTILE * (KC / 4); i += 256) {
            int r  = i >> 3;          // / (KC/4)
            int c4 = i & 7;
            float4 val = *(const float4*)(x + (rowBlock + r) * CD + kc + c4 * 4);
            float* dst = &xs[r * XPAD + c4 * 4];
            dst[0] = val.x; dst[1] = val.y; dst[2] = val.z; dst[3] = val.w;
        }
        // --- stage W chunks transposed: ws[m][n*XPAD + k] = W[kc+k][n] ---
        #pragma unroll
        for (int m = 0; m < 3; ++m) {
            for (int i = tid; i < KC * HD; i += 256) {
                int k = i >> 6;       // 0..31
                int n = i & 63;       // 0..63
                ws[m][n * XPAD + k] = Wm[m][(long)(kc + k) * HD + n];
            }
        }
        // prefetch next x chunk while we compute on this one
        if (kc + KC < CD) {
            const float* nx = x + (rowBlock + (tid >> 1)) * CD + kc + KC + (tid & 1) * 16;
            __builtin_prefetch(nx, 0, 1);
        }
        __syncthreads();

        const float* xw = &xs[(wave * 16 + idx) * XPAD];
        #pragma unroll
        for (int kk = 0; kk < KC; kk += 4) {
            v2f a = *(const v2f*)(xw + kk + 2 * grp);   // A[m=idx][kk+2g, kk+2g+1]
            #pragma unroll
            for (int m = 0; m < 3; ++m) {
                #pragma unroll
                for (int t = 0; t < 4; ++t) {
                    v2f b = *(const v2f*)(&ws[m][(t * 16 + idx) * XPAD + kk + 2 * grp]);
                    acc[m][t] = wmma_f32(a, b, acc[m][t]);
                }
            }
        }
    }

    // --- store: C-layout VGPR r -> row (r + 8*grp), col = t*16 + idx ---
    const long rowW = rowBlock + wave * 16;
    #pragma unroll
    for (int m = 0; m < 3; ++m) {
        float* Om = (m == 0) ? Q : ((m == 1) ? K : V);
        #pragma unroll
        for (int t = 0; t < 4; ++t)
            #pragma unroll
            for (int r = 0; r < 8; ++r)
                Om[(rowW + r + 8 * grp) * HD + t * 16 + idx] = acc[m][t][r];
    }
}

// ---------------------------------------------------------------------------
// Kernel 2: causal flash attention.
// grid = (T/128, B) x 256 threads. Wave w owns 16 query rows; K/V chunks of
// 64 keys staged in LDS (V transposed); online softmax in C-layout.
// ---------------------------------------------------------------------------
__global__ __launch_bounds__(256) void attn_kernel(
    const float* __restrict__ Q, const float* __restrict__ K,
    const float* __restrict__ V, float* __restrict__ out)
{
    __shared__ float ks[SK * KPAD];        // K chunk: [key][hd] (+pad)
    __shared__ float vs[HD * KPAD];        // V chunk transposed: [hd][key] (+pad)
    __shared__ float ps[8][16 * 16];       // per-wave P tile staging

    const int tid  = threadIdx.x;
    const int wave = tid >> 5;
    const int lane = tid & 31;
    const int grp  = lane >> 4;
    const int idx  = lane & 15;

    const int  b     = blockIdx.y;
    const int  qt    = blockIdx.x;               // query tile 0..15
    const long base  = (long)b * TT * HD;
    const int  qrow0 = qt * MTILE + wave * 16;   // this wave's first query row

    // --- register-resident Q fragments: qf[s] = Q[qrow0+idx][4s + 2g .. +1] ---
    v2f qf[16];
    const float* qp = Q + base + (long)(qrow0 + idx) * HD + 2 * grp;
    #pragma unroll
    for (int s = 0; s < 16; ++s) qf[s] = *(const v2f*)(qp + 4 * s);

    v8f   o[4] = {};                             // 16x64 output accumulator
    float mrow[8], lrow[8];
    #pragma unroll
    for (int r = 0; r < 8; ++r) { mrow[r] = -1e30f; lrow[r] = 0.f; }

    const int nchunk = (qt + 1) * MTILE / SK;    // causal: keys <= qt*128+127
    for (int sc = 0; sc < nchunk; ++sc) {
        const int s0 = sc * SK;
        __syncthreads();
        // --- stage K chunk and transposed V chunk ---
        for (int i = tid; i < SK * (HD / 4); i += 256) {
            int r  = i >> 4;                      // key row 0..63
            int c4 = i & 15;                      // float4 column
            float4 kv = *(const float4*)(K + base + (long)(s0 + r) * HD + c4 * 4);
            float* dst = &ks[r * KPAD + c4 * 4];
            dst[0] = kv.x; dst[1] = kv.y; dst[2] = kv.z; dst[3] = kv.w;
            float4 vv = *(const float4*)(V + base + (long)(s0 + r) * HD + c4 * 4);
            vs[(c4 * 4 + 0) * KPAD + r] = vv.x;
            vs[(c4 * 4 + 1) * KPAD + r] = vv.y;
            vs[(c4 * 4 + 2) * KPAD + r] = vv.z;
            vs[(c4 * 4 + 3) * KPAD + r] = vv.w;
        }
        if (sc + 1 < nchunk) {
            const long nrow = base + (long)(s0 + SK + (tid >> 2)) * HD + (tid & 3) * 16;
            __builtin_prefetch(K + nrow, 0, 1);
            __builtin_prefetch(V + nrow, 0, 1);
        }
        __syncthreads();

        #pragma unroll
        for (int kt = 0; kt < 4; ++kt) {
            // --- scores S(16q x 16k) = Q(16x64) x K^T(64x16) ---
            v8f s = {};
            const float* kw = &ks[(kt * 16 + idx) * KPAD + 2 * grp];
            #pragma unroll
            for (int st = 0; st < 16; ++st) {
                v2f bfrag = *(const v2f*)(kw + 4 * st);  // B[k][n]=K[key=n][k]
                s = wmma_f32(qf[st], bfrag, s);
            }

            const int key = s0 + kt * 16 + idx;          // this lane's key column
            // scale + causal mask
            #pragma unroll
            for (int r = 0; r < 8; ++r) {
                const int qr = qrow0 + r + 8 * grp;
                float sv = s[r] * 0.125f;                // H^-0.5 = 1/8
                s[r] = (key <= qr) ? sv : -1e30f;
            }
            // row max over the 16 lanes holding this row
            float tmax[8];
            #pragma unroll
            for (int r = 0; r < 8; ++r) {
                float v = s[r];
                #pragma unroll
                for (int off = 1; off < 16; off <<= 1)
                    v = fmaxf(v, __shfl_xor(v, off, 32));
                tmax[r] = v;
            }
            // online softmax update
            float pcorr[8];
            #pragma unroll
            for (int r = 0; r < 8; ++r) {
                const float mnew = fmaxf(mrow[r], tmax[r]);
                pcorr[r] = __expf(mrow[r] - mnew);
                mrow[r]  = mnew;
                const int qr = qrow0 + r + 8 * grp;
                s[r] = (key <= qr) ? __expf(s[r] - mnew) : 0.f;  // hard-zero masked
            }
            // row sums -> l
            #pragma unroll
            for (int r = 0; r < 8; ++r) {
                float v = s[r];
                #pragma unroll
                for (int off = 1; off < 16; off <<= 1)
                    v += __shfl_xor(v, off, 32);
                lrow[r] = lrow[r] * pcorr[r] + v;
            }
            // rescale running output
            #pragma unroll
            for (int t = 0; t < 4; ++t)
                #pragma unroll
                for (int r = 0; r < 8; ++r)
                    o[t][r] *= pcorr[r];

            // --- C-layout -> A-layout for P via per-wave LDS patch ---
            float* pw = &ps[wave][0];
            #pragma unroll
            for (int r = 0; r < 8; ++r)
                pw[(r + 8 * grp) * 16 + idx] = s[r];
            // same-wave DS ordering: compiler inserts s_wait_dscnt before reads

            // --- O(16x64) += P(16x16) x V(16x64) ---
            #pragma unroll
            for (int pk = 0; pk < 16; pk += 4) {
                v2f a = *(const v2f*)(&pw[idx * 16 + pk + 2 * grp]);
                #pragma unroll
                for (int t = 0; t < 4; ++t) {
                    v2f bfrag = *(const v2f*)(&vs[(t * 16 + idx) * KPAD + kt * 16 + pk + 2 * grp]);
                    o[t] = wmma_f32(a, bfrag, o[t]);
                }
            }
        }
    }

    // --- normalize and store ---
    #pragma unroll
    for (int t = 0; t < 4; ++t)
        #pragma unroll
        for (int r = 0; r < 8; ++r) {
            const long row = qrow0 + r + 8 * grp;
            out[base + row * HD + t * 16 + idx] = o[t][r] / lrow[r];
        }
}

// ---------------------------------------------------------------------------
extern "C" void kernel_launch(void* const* d_in, const int* in_sizes, int n_in,
                              void* d_out, int out_size, void* d_ws, size_t ws_size,
                              hipStream_t stream) {
    const float* x  = (const float*)d_in[0];
    const float* Wq = (const float*)d_in[1];
    const float* Wk = (const float*)d_in[2];
    const float* Wv = (const float*)d_in[3];
    float* out = (float*)d_out;

    // workspace: Q | K | V, each BT*HD f32 (4 MB) -> 12 MB total
    float* Q = (float*)d_ws;
    float* K = Q + (size_t)BT * HD;
    float* V = K + (size_t)BT * HD;

    proj_qkv_kernel<<<dim3(BT / MTILE), 256, 0, stream>>>(x, Wq, Wk, Wv, Q, K, V);
    attn_kernel<<<dim3(TT / MTILE, BB), 256, 0, stream>>>(Q, K, V, out);
}